// KLDLoss_85796266704897
// MI455X (gfx1250) — compile-verified
//
#include <hip/hip_runtime.h>
#include <stdint.h>

// ---------------------------------------------------------------------------
// Segmented stable-logsumexp KLD loss, MI455X (gfx1250).
//   Kernel 1 (per 8192-elem block tile):
//     - attn tile: global -> LDS via global_load_async_to_lds_b128 (async DMA),
//       then one ds_load_b128 sweep into registers (reused by both passes).
//     - cluster ids: global int4 loads, packed 4 x u8 per VGPR (no LDS).
//     - pass 1: per-cluster {count, sum_a, max} via LDS atomics (SoA, bank c).
//     - pass 2: per-cluster sum exp(a - max_c) via ds_add_f32.
//   Kernel 2: merge per-block partials with stable (m,s) logsumexp combine.
// ---------------------------------------------------------------------------

#define KMAX   64
#define TPB    256
#define CHUNK  8192                    // elements per block tile
#define VPT    (CHUNK / TPB)           // 32 elements per thread
#define LOADS  (VPT / 4)               // 8 x b128 per thread

__device__ __forceinline__ unsigned enc_f32(float f) {
  unsigned u = __float_as_uint(f);
  return (u & 0x80000000u) ? ~u : (u | 0x80000000u);   // monotone f32 -> u32
}
__device__ __forceinline__ float dec_f32(unsigned k) {
  unsigned u = (k & 0x80000000u) ? (k & 0x7fffffffu) : ~k;
  return __uint_as_float(u);
}

// LDS byte offset of a __shared__ object: low 32 bits of the generic address
// are the wave-relative LDS address (ISA 10.2 aperture mapping).
__device__ __forceinline__ unsigned lds_off(const void* p) {
  return (unsigned)(uintptr_t)p;
}

__global__ __launch_bounds__(TPB) void seg_part_kernel(
    const float* __restrict__ a, const int* __restrict__ cl,
    float4* __restrict__ partial, int n) {
  __shared__ __align__(16) float a_buf[CHUNK];
  __shared__ unsigned s_cnt[KMAX];
  __shared__ unsigned s_maxk[KMAX];
  __shared__ float    s_maxf[KMAX];
  __shared__ float    s_se[KMAX];
  __shared__ float    s_sa[KMAX];

  const int tid = threadIdx.x;
  const long long base = (long long)blockIdx.x * CHUNK;

  if (tid < KMAX) {
    s_cnt[tid]  = 0u;
    s_maxk[tid] = 0u;     // below enc of any real value
    s_se[tid]   = 0.0f;
    s_sa[tid]   = 0.0f;
  }

  float4   av[LOADS];     // 32 attn values in registers
  unsigned cp[LOADS];     // 32 cluster ids, 4 x u8 packed per reg

  const bool full = (base + CHUNK) <= (long long)n;   // uniform per block
  if (full) {
    // Async DMA the attn tile into LDS (16 B/lane, fully coalesced),
    // and in parallel pull cluster ids straight into registers.
#pragma unroll
    for (int j = 0; j < LOADS; ++j) {
      const int e = (j * TPB + tid) * 4;          // element index in tile
      const float* ga = a + base + e;
      unsigned la = lds_off(&a_buf[e]);
      asm volatile("global_load_async_to_lds_b128 %0, %1, off"
                   :: "v"(la), "v"(ga) : "memory");
    }
    const int4* c4 = (const int4*)(cl + base);
#pragma unroll
    for (int j = 0; j < LOADS; ++j) {
      const int4 ci = c4[j * TPB + tid];
      cp[j] = (unsigned)(ci.x & 0xff) | ((unsigned)(ci.y & 0xff) << 8) |
              ((unsigned)(ci.z & 0xff) << 16) | ((unsigned)(ci.w & 0xff) << 24);
    }
    asm volatile("s_wait_asynccnt 0" ::: "memory");
  } else {
    // Tail tile (not taken for N = 8388608): bounds-checked scalar loads.
#pragma unroll
    for (int j = 0; j < LOADS; ++j) {
      float v[4];
      unsigned p = 0;
#pragma unroll
      for (int q = 0; q < 4; ++q) {
        const long long g = base + (long long)(j * TPB + tid) * 4 + q;
        if (g < (long long)n) {
          v[q] = a[g];
          p |= ((unsigned)(cl[g] & 0xff)) << (8 * q);
        } else {
          v[q] = 0.0f;
          p |= 0xffu << (8 * q);                  // sentinel: skipped below
        }
      }
      av[j] = make_float4(v[0], v[1], v[2], v[3]);
      cp[j] = p;
    }
  }
  __syncthreads();   // accumulator init + all waves' async LDS writes visible

  if (full) {
    // One conflict-optimal b128 sweep of the staged tile into registers.
    const float4* ab4 = (const float4*)a_buf;
#pragma unroll
    for (int j = 0; j < LOADS; ++j) av[j] = ab4[j * TPB + tid];
  }

  // Pass 1: count, sum(a), max(a) per cluster.
#pragma unroll
  for (int j = 0; j < LOADS; ++j) {
    const unsigned cc = cp[j];
    const float4 v = av[j];
#pragma unroll
    for (int q = 0; q < 4; ++q) {
      const unsigned c = (cc >> (8 * q)) & 0xffu;
      if (c < KMAX) {
        const float x = (q == 0) ? v.x : (q == 1) ? v.y : (q == 2) ? v.z : v.w;
        atomicAdd(&s_cnt[c], 1u);
        atomicAdd(&s_sa[c], x);
        atomicMax(&s_maxk[c], enc_f32(x));
      }
    }
  }
  __syncthreads();
  if (tid < KMAX) s_maxf[tid] = dec_f32(s_maxk[tid]);
  __syncthreads();

  // Pass 2: sum exp(a - max_c), data already in registers.
#pragma unroll
  for (int j = 0; j < LOADS; ++j) {
    const unsigned cc = cp[j];
    const float4 v = av[j];
#pragma unroll
    for (int q = 0; q < 4; ++q) {
      const unsigned c = (cc >> (8 * q)) & 0xffu;
      if (c < KMAX) {
        const float x = (q == 0) ? v.x : (q == 1) ? v.y : (q == 2) ? v.z : v.w;
        atomicAdd(&s_se[c], __expf(x - s_maxf[c]));
      }
    }
  }
  __syncthreads();

  if (tid < KMAX) {
    const unsigned cn = s_cnt[tid];
    float4 p;
    p.x = (float)cn;
    p.y = cn ? s_maxf[tid] : 0.0f;   // keep finite; skipped in merge if cn==0
    p.z = s_se[tid];
    p.w = s_sa[tid];
    partial[(size_t)blockIdx.x * KMAX + tid] = p;
  }
}

__global__ __launch_bounds__(64) void seg_final_kernel(
    const float4* __restrict__ partial, int nblocks,
    const int* __restrict__ kptr, float* __restrict__ out) {
  __shared__ float red[KMAX];
  const int c = threadIdx.x;          // one cluster per thread (64 = 2 waves)
  const int K = *kptr;

  float cnt = 0.0f, m = -3.402823466e38f, se = 0.0f, sa = 0.0f;
  for (int b = 0; b < nblocks; ++b) {
    const float4 p = partial[(size_t)b * KMAX + c];
    if (p.x > 0.0f) {
      cnt += p.x;
      sa  += p.w;
      if (p.y > m) {                  // stable (m, s) logsumexp combine
        se = se * __expf(m - p.y) + p.z;
        m  = p.y;
      } else {
        se += p.z * __expf(p.y - m);
      }
    }
  }

  float loss = 0.0f;
  if (c < K && cnt > 4.0f) {
    const float lse = m + logf(fmaxf(se, 1e-38f));
    loss = -logf(cnt) - (sa - cnt * lse) / cnt;
  }
  red[c] = loss;
  __syncthreads();
  if (c == 0) {
    float t = 0.0f;
    for (int i = 0; i < KMAX; ++i) t += red[i];
    out[0] = t;
  }
}

extern "C" void kernel_launch(void* const* d_in, const int* in_sizes, int n_in,
                              void* d_out, int out_size, void* d_ws, size_t ws_size,
                              hipStream_t stream) {
  const float* a    = (const float*)d_in[0];   // attn_val, [1, N] f32
  const int*   cl   = (const int*)d_in[1];     // cluster,  [N] int
  const int*   kptr = (const int*)d_in[2];     // num_clusters scalar
  const int n = in_sizes[0];

  const int nblocks = (n + CHUNK - 1) / CHUNK; // 1024 for N = 8388608
  float4* partial = (float4*)d_ws;             // nblocks*64*16 B = 1 MB

  seg_part_kernel<<<nblocks, TPB, 0, stream>>>(a, cl, partial, n);
  seg_final_kernel<<<1, 64, 0, stream>>>(partial, nblocks, kptr, (float*)d_out);
}